// BertSelfAttention_27556510171402
// MI455X (gfx1250) — compile-verified
//
#include <hip/hip_runtime.h>
#include <hip/hip_bf16.h>
#include <cmath>

// ---------------------------------------------------------------------------
// BERT self-attention for MI455X (gfx1250, wave32, WMMA + async-to-LDS).
// B=4, S=2048, D=1024, H=16, DK=64.  All math in f32 via V_WMMA_F32_16X16X4_F32
// (memory-bound: attn output alone is ~1.07 GB -> ~46us HBM floor; f32 WMMA
// dense throughput is orders of magnitude above the needed ~3 TFLOPS).
// GEMM tile staging uses GLOBAL_LOAD_ASYNC_TO_LDS_B128 with double buffering.
// ---------------------------------------------------------------------------

typedef float v2f __attribute__((ext_vector_type(2)));
typedef float v8f __attribute__((ext_vector_type(8)));
typedef int   v4i __attribute__((ext_vector_type(4)));

#define BB   4
#define SS   2048
#define DD   1024
#define HH   16
#define DKK  64
#define MM   (BB * SS)          // 8192 rows for the projections

__device__ __forceinline__ v8f wmma4(v2f a, v2f b, v8f c) {
    // D(16x16,f32) = A(16x4,f32) * B(4x16,f32) + C
    return __builtin_amdgcn_wmma_f32_16x16x4_f32(
        /*neg_a=*/false, a, /*neg_b=*/false, b,
        /*c_mod=*/(short)0, c, /*reuse_a=*/false, /*reuse_b=*/false);
}

// ---- CDNA5 async copy: 16B global -> LDS, tracked by ASYNCcnt -------------
#if defined(__has_builtin) && __has_builtin(__builtin_amdgcn_global_load_async_to_lds_b128)
typedef __attribute__((address_space(1))) v4i gas_v4i;   // global
typedef __attribute__((address_space(3))) v4i las_v4i;   // LDS
__device__ __forceinline__ void async_copy16(const float* g, float* l) {
    __builtin_amdgcn_global_load_async_to_lds_b128(
        (gas_v4i*)g, (las_v4i*)l, /*offset=*/0, /*cpol=*/0);
}
#else
__device__ __forceinline__ void async_copy16(const float* g, float* l) {
    unsigned loff =
        (unsigned)(size_t)(__attribute__((address_space(3))) void*)l;
    asm volatile("global_load_async_to_lds_b128 %0, %1, off"
                 :: "v"(loff), "v"(g) : "memory");
}
#endif

__device__ __forceinline__ void wait_async0() {
#if defined(__has_builtin) && __has_builtin(__builtin_amdgcn_s_wait_asynccnt)
    __builtin_amdgcn_s_wait_asynccnt(0);
#else
    asm volatile("s_wait_asynccnt 0x0" ::: "memory");
#endif
}

// ---------------------------------------------------------------------------
// Kernel 1: fused QKV projection.  Y = X @ W^T + bias, scattered head-major
// into [B,H,S,DK].  blockIdx.z in {0,1,2} selects (Wq,bq,Q)/(Wk,..)/(Wv,..).
// 256 threads (8 waves), 64x64 output tile, K-chunks of 16, double-buffered
// async-to-LDS staging.
// ---------------------------------------------------------------------------
#define LSTR 20   // LDS row stride (floats): 16B-aligned rows, conflict-free

__global__ __launch_bounds__(256)
void qkv_gemm_kernel(const float* __restrict__ X,
                     const float* __restrict__ W0, const float* __restrict__ b0,
                     const float* __restrict__ W1, const float* __restrict__ b1,
                     const float* __restrict__ W2, const float* __restrict__ b2,
                     float* __restrict__ Y0, float* __restrict__ Y1,
                     float* __restrict__ Y2)
{
    const int mz = blockIdx.z;
    const float* W    = (mz == 0) ? W0 : ((mz == 1) ? W1 : W2);
    const float* bias = (mz == 0) ? b0 : ((mz == 1) ? b1 : b2);
    float*       Y    = (mz == 0) ? Y0 : ((mz == 1) ? Y1 : Y2);

    const int m0 = blockIdx.x * 64;
    const int e0 = blockIdx.y * 64;

    __shared__ float Xs[2][64 * LSTR];
    __shared__ float Ws[2][64 * LSTR];

    const int tid  = threadIdx.x;
    const int lane = tid & 31;
    const int wv   = tid >> 5;
    const int l16  = lane & 15;
    const int wm   = wv >> 1;        // 0..3 : M sub-tile
    const int wn2  = wv & 1;         // 0..1 : pair of N sub-tiles
    const int koff = (lane < 16) ? 0 : 2;

    v8f acc0 = {};
    v8f acc1 = {};

    const int srow = tid >> 2;
    const int scol = (tid & 3) * 4;
    const int sidx = srow * LSTR + scol;          // 16B aligned (LSTR*4=80)
    const float* Xg = X + (size_t)(m0 + srow) * DD + scol;
    const float* Wg = W + (size_t)(e0 + srow) * DD + scol;

    // prologue: stage chunk 0
    async_copy16(Xg, &Xs[0][sidx]);
    async_copy16(Wg, &Ws[0][sidx]);
    wait_async0();
    __syncthreads();

    for (int kb = 0; kb < DD; kb += 16) {
        const int cur = (kb >> 4) & 1;
        const int nxt = cur ^ 1;
        if (kb + 16 < DD) {                       // stage ahead (async engine)
            async_copy16(Xg + kb + 16, &Xs[nxt][sidx]);
            async_copy16(Wg + kb + 16, &Ws[nxt][sidx]);
        }

#pragma unroll
        for (int kk = 0; kk < 4; ++kk) {
            const int k = kk * 4 + koff;
            v2f a  = *(const v2f*)(&Xs[cur][(wm * 16 + l16) * LSTR + k]);
            v2f ba = *(const v2f*)(&Ws[cur][(wn2 * 32 + l16) * LSTR + k]);
            v2f bb = *(const v2f*)(&Ws[cur][(wn2 * 32 + 16 + l16) * LSTR + k]);
            acc0 = wmma4(a, ba, acc0);
            acc1 = wmma4(a, bb, acc1);
        }

        wait_async0();                            // next buffer landed
        __syncthreads();                          // everyone done with cur
    }

#pragma unroll
    for (int t = 0; t < 2; ++t) {
        v8f acc = t ? acc1 : acc0;
        const int e  = e0 + wn2 * 32 + t * 16 + l16;
        const float bv = bias[e];
        const int h  = e >> 6;
        const int dk = e & 63;
#pragma unroll
        for (int r = 0; r < 8; ++r) {
            const int m    = m0 + wm * 16 + r + ((lane >= 16) ? 8 : 0);
            const int bidx = m >> 11;           // m / S
            const int s    = m & (SS - 1);      // m % S
            Y[(((size_t)(bidx * HH + h)) * SS + s) * DKK + dk] = acc[r] + bv;
        }
    }
}

// ---------------------------------------------------------------------------
// Kernel 2: fused attention for one (b, h, 16-row q-tile).
//   Phase 1: scores(16x2048) = Q Kt * 0.125, masked -> LDS
//   Phase 2: exact softmax per row; write attn once; normalize LDS in place
//   Phase 3: ctx(16x64) = P @ V, split-K across 8 waves, LDS combine
// LDS: 16*2052 score floats (131 KB) + 4*16*17 partials  (320KB WGP LDS).
// ---------------------------------------------------------------------------
#define SCS 2052   // score row stride: even (float2-aligned), good bank spread

__global__ __launch_bounds__(256)
void attention_kernel(const float* __restrict__ Q, const float* __restrict__ K,
                      const float* __restrict__ V, const int* __restrict__ mask,
                      float* __restrict__ attnOut, float* __restrict__ ctx)
{
    extern __shared__ float smem[];
    float* Sc   = smem;                 // 16 * SCS
    float* part = smem + 16 * SCS;      // 4 * 16 * 17

    const int qt = blockIdx.x;          // 0..127
    const int h  = blockIdx.y;          // 0..15
    const int b  = blockIdx.z;          // 0..3
    const int bh = b * HH + h;
    const int m0 = qt * 16;

    const int tid  = threadIdx.x;
    const int lane = tid & 31;
    const int wv   = tid >> 5;
    const int l16  = lane & 15;
    const int koff = (lane < 16) ? 0 : 2;

    // ---- Phase 1: scores -------------------------------------------------
    // Preload the A operand (Q rows m0..m0+15, full DK=64) into registers:
    // 16 K-chunks of 4 -> 16 float2 per lane (WMMA 16x4 f32 A layout).
    const float* Qrow = Q + ((size_t)bh * SS + m0 + l16) * DKK;
    v2f a_reg[16];
#pragma unroll
    for (int kc = 0; kc < 16; ++kc)
        a_reg[kc] = *(const v2f*)(Qrow + kc * 4 + koff);

    const float* Kb = K + (size_t)bh * SS * DKK;
    for (int t = 0; t < 16; ++t) {              // 16 key tiles per wave
        const int key0 = wv * 256 + t * 16;
        v8f acc = {};
#pragma unroll
        for (int kc = 0; kc < 16; ++kc) {
            // B(4x16) = K^T chunk: lane n=key, rows dk (float2 global load)
            v2f bb = *(const v2f*)(Kb + (size_t)(key0 + l16) * DKK + kc * 4 + koff);
            acc = wmma4(a_reg[kc], bb, acc);
        }
        const int key     = key0 + l16;
        const bool masked = (mask[b * SS + key] == 0);
#pragma unroll
        for (int r = 0; r < 8; ++r) {
            const int row = r + ((lane >= 16) ? 8 : 0);
            float vsc = acc[r] * 0.125f;        // 1/sqrt(64)
            if (masked) vsc = -INFINITY;
            Sc[row * SCS + key] = vsc;
        }
    }
    __syncthreads();

    // ---- Phase 2: softmax ------------------------------------------------
    {
        const int row = 2 * wv + ((lane >= 16) ? 1 : 0);   // 8 waves x 2 rows
        float* rowp = Sc + row * SCS;
        float mx = -INFINITY;
        for (int i = 0; i < 128; ++i) mx = fmaxf(mx, rowp[l16 + 16 * i]);
#pragma unroll
        for (int off = 8; off; off >>= 1) mx = fmaxf(mx, __shfl_xor(mx, off, 16));
        float sum = 0.0f;
        for (int i = 0; i < 128; ++i) sum += __expf(rowp[l16 + 16 * i] - mx);
#pragma unroll
        for (int off = 8; off; off >>= 1) sum += __shfl_xor(sum, off, 16);
        const float inv = 1.0f / sum;

        float* arow = attnOut + ((size_t)bh * SS + m0 + row) * SS;
        for (int i = 0; i < 128; ++i) {
            const int c   = l16 + 16 * i;
            const float p = __expf(rowp[c] - mx) * inv;
            rowp[c] = p;        // keep normalized probs in LDS for phase 3
            arow[c] = p;        // the only HBM write of attn (1.07 GB total)
        }
    }
    __syncthreads();

    // ---- Phase 3: ctx = P @ V -------------------------------------------
    const int ntile = wv & 3;       // dk tile 0..3
    const int khalf = wv >> 2;      // split-K half
    const int dk0   = ntile * 16;
    const float* Vb = V + (size_t)bh * SS * DKK;

    v8f acc = {};
    for (int kc = 0; kc < 256; ++kc) {
        const int k = khalf * 1024 + kc * 4 + koff;
        if ((kc & 7) == 0)          // keep the V stream ahead in cache
            __builtin_prefetch(Vb + (size_t)(k + 32) * DKK + dk0 + l16, 0, 1);
        v2f a = *(const v2f*)(Sc + l16 * SCS + k);   // P 16x4 from LDS
        v2f bb;
        bb.x = Vb[(size_t)k * DKK + dk0 + l16];      // V rows k, k+1
        bb.y = Vb[(size_t)(k + 1) * DKK + dk0 + l16];
        acc = wmma4(a, bb, acc);
    }

    if (khalf == 1) {
#pragma unroll
        for (int r = 0; r < 8; ++r) {
            const int row = r + ((lane >= 16) ? 8 : 0);
            part[(ntile * 16 + row) * 17 + l16] = acc[r];
        }
    }
    __syncthreads();
    if (khalf == 0) {
#pragma unroll
        for (int r = 0; r < 8; ++r) {
            const int row = r + ((lane >= 16) ? 8 : 0);
            const float val = acc[r] + part[(ntile * 16 + row) * 17 + l16];
            const int s = m0 + row;
            // ctx stored [B,S,D] so the output projection reads row-major
            ctx[((size_t)b * SS + s) * DD + h * DKK + dk0 + l16] = val;
        }
    }
}

// ---------------------------------------------------------------------------
// Kernel 3: output projection.  out = ctx @ Wo^T + bo, row-major [M, D].
// Same structure as kernel 1 (async double-buffered staging).
// ---------------------------------------------------------------------------
__global__ __launch_bounds__(256)
void out_gemm_kernel(const float* __restrict__ X, const float* __restrict__ W,
                     const float* __restrict__ bias, float* __restrict__ Y)
{
    const int m0 = blockIdx.x * 64;
    const int e0 = blockIdx.y * 64;

    __shared__ float Xs[2][64 * LSTR];
    __shared__ float Ws[2][64 * LSTR];

    const int tid  = threadIdx.x;
    const int lane = tid & 31;
    const int wv   = tid >> 5;
    const int l16  = lane & 15;
    const int wm   = wv >> 1;
    const int wn2  = wv & 1;
    const int koff = (lane < 16) ? 0 : 2;

    v8f acc0 = {};
    v8f acc1 = {};

    const int srow = tid >> 2;
    const int scol = (tid & 3) * 4;
    const int sidx = srow * LSTR + scol;
    const float* Xg = X + (size_t)(m0 + srow) * DD + scol;
    const float* Wg = W + (size_t)(e0 + srow) * DD + scol;

    async_copy16(Xg, &Xs[0][sidx]);
    async_copy16(Wg, &Ws[0][sidx]);
    wait_async0();
    __syncthreads();

    for (int kb = 0; kb < DD; kb += 16) {
        const int cur = (kb >> 4) & 1;
        const int nxt = cur ^ 1;
        if (kb + 16 < DD) {
            async_copy16(Xg + kb + 16, &Xs[nxt][sidx]);
            async_copy16(Wg + kb + 16, &Ws[nxt][sidx]);
        }

#pragma unroll
        for (int kk = 0; kk < 4; ++kk) {
            const int k = kk * 4 + koff;
            v2f a  = *(const v2f*)(&Xs[cur][(wm * 16 + l16) * LSTR + k]);
            v2f ba = *(const v2f*)(&Ws[cur][(wn2 * 32 + l16) * LSTR + k]);
            v2f bb = *(const v2f*)(&Ws[cur][(wn2 * 32 + 16 + l16) * LSTR + k]);
            acc0 = wmma4(a, ba, acc0);
            acc1 = wmma4(a, bb, acc1);
        }

        wait_async0();
        __syncthreads();
    }

#pragma unroll
    for (int t = 0; t < 2; ++t) {
        v8f acc = t ? acc1 : acc0;
        const int e = e0 + wn2 * 32 + t * 16 + l16;
        const float bv = bias[e];
#pragma unroll
        for (int r = 0; r < 8; ++r) {
            const int m = m0 + wm * 16 + r + ((lane >= 16) ? 8 : 0);
            Y[(size_t)m * DD + e] = acc[r] + bv;
        }
    }
}

// ---------------------------------------------------------------------------
extern "C" void kernel_launch(void* const* d_in, const int* in_sizes, int n_in,
                              void* d_out, int out_size, void* d_ws, size_t ws_size,
                              hipStream_t stream)
{
    const float* hs   = (const float*)d_in[0];
    const int*   mask = (const int*)d_in[1];
    const float* Wq   = (const float*)d_in[2];
    const float* bq   = (const float*)d_in[3];
    const float* Wk   = (const float*)d_in[4];
    const float* bk   = (const float*)d_in[5];
    const float* Wv   = (const float*)d_in[6];
    const float* bv   = (const float*)d_in[7];
    const float* Wo   = (const float*)d_in[8];
    const float* bo   = (const float*)d_in[9];

    float* out  = (float*)d_out;                               // [B,S,D]
    float* attn = out + (size_t)BB * SS * DD;                  // [B,H,S,S]

    const size_t qkv_elems = (size_t)BB * HH * SS * DKK;       // 8.4M floats
    float* Qb  = (float*)d_ws;
    float* Kb  = Qb + qkv_elems;
    float* Vb  = Kb + qkv_elems;
    float* ctx = Vb + qkv_elems;                               // [B,S,D]

    // 1) QKV projections (one launch, z selects Q/K/V)
    dim3 g1(MM / 64, DD / 64, 3);
    qkv_gemm_kernel<<<g1, 256, 0, stream>>>(hs, Wq, bq, Wk, bk, Wv, bv,
                                            Qb, Kb, Vb);

    // 2) fused attention (131 KB score tile in the 320 KB WGP LDS)
    const size_t smem2 = (size_t)(16 * SCS + 4 * 16 * 17) * sizeof(float);
    (void)hipFuncSetAttribute((const void*)attention_kernel,
                              hipFuncAttributeMaxDynamicSharedMemorySize,
                              (int)smem2);
    dim3 g2(SS / 16, HH, BB);
    attention_kernel<<<g2, 256, smem2, stream>>>(Qb, Kb, Vb, mask, attn, ctx);

    // 3) output projection
    dim3 g3(MM / 64, DD / 64, 1);
    out_gemm_kernel<<<g3, 256, 0, stream>>>(ctx, Wo, bo, out);
}